// AngularComponent_19507741458667
// MI455X (gfx1250) — compile-verified
//
#include <hip/hip_runtime.h>
#include <stdint.h>

#define BLOCK 256
#define K 20   // monomial count for l_max = 3

// Streaming monomial kernel.  Store-bandwidth bound: 160 MB out + 24 MB in
// => ~7.9 us floor at 23.3 TB/s.  16 VALU mults/point, no matrix structure,
// so the CDNA5 feature used is the async LDS->global store engine for
// perfectly coalesced 512 B/instruction output stores.
__global__ __launch_bounds__(BLOCK) void angular_monomials_kernel(
    const float* __restrict__ vec, float* __restrict__ out, int n)
{
    // Staging buffer mirrors this block's output region: 256 rows x 80 B.
    __shared__ float buf[BLOCK * K];

    const int tid  = (int)threadIdx.x;
    const int lane = tid & 31;
    const int wave = tid >> 5;
    const long long blockBase = (long long)blockIdx.x * BLOCK;
    const long long p = blockBase + tid;

    float x = 0.f, y = 0.f, z = 0.f;
    if (p < (long long)n) {
        const float* v = vec + 3 * p;   // contiguous 12 B/lane -> coalesced wave read
        x = v[0]; y = v[1]; z = v[2];
    }

    // Shared subexpressions: 6 + 10 = 16 multiplies total.
    const float x2 = x * x, y2 = y * y, z2 = z * z;
    const float xy = x * y, xz = x * z, yz = y * z;

    // Stage this thread's 20 outputs (enumeration order) as 5x ds_store_b128.
    // Row offset tid*80 B is 16-byte aligned.
    float4* lrow = (float4*)&buf[tid * K];
    lrow[0] = make_float4(1.0f,    x,       y,       z);
    lrow[1] = make_float4(x2,      xy,      xz,      y2);
    lrow[2] = make_float4(yz,      z2,      x2 * x,  x2 * y);
    lrow[3] = make_float4(x2 * z,  y2 * x,  xy * z,  z2 * x);
    lrow[4] = make_float4(y2 * y,  y2 * z,  z2 * y,  z2 * z);

    // Intra-wave only: the async stores below read back exactly this wave's
    // 32 rows, so a DScnt drain (no barrier) orders LDS writes vs. the DMA.
    asm volatile("s_wait_dscnt 0x0" ::: "memory");

    // Each wave drains its own 2560 B region, 16 B chunks, lane-contiguous:
    // store i: lane l writes chunk c = i*32+l  ->  512 B coalesced per issue.
    const long long waveBase = blockBase + (long long)(wave * 32);
    const long long rem = (long long)n - waveBase;
    const int validPts = rem >= 32 ? 32 : (rem > 0 ? (int)rem : 0);

    const uint64_t gbase = (uint64_t)(uintptr_t)(out + waveBase * K);
    const unsigned lbase = (unsigned)(uintptr_t)(void*)&buf[wave * 32 * K];

    #pragma unroll
    for (int i = 0; i < 5; ++i) {
        const int c = i * 32 + lane;            // 16 B chunk id in wave region
        if (c < validPts * 5) {                 // 5 chunks per point, never straddles
            uint64_t gaddr = gbase + (uint64_t)c * 16u;
            unsigned laddr = lbase + (unsigned)c * 16u;
            // GV mode: 64-bit address in VGPR pair, LDS byte addr in VSRC.
            asm volatile("global_store_async_from_lds_b128 %0, %1, off"
                         :: "v"(gaddr), "v"(laddr)
                         : "memory");
        }
    }

    // Drain ASYNCcnt before the wave retires (S_ENDPGM also wait-idles, but
    // be explicit so LDS lifetime can never end under outstanding DMA).
    asm volatile("s_wait_asynccnt 0x0" ::: "memory");
}

extern "C" void kernel_launch(void* const* d_in, const int* in_sizes, int n_in,
                              void* d_out, int out_size, void* d_ws, size_t ws_size,
                              hipStream_t stream)
{
    (void)n_in; (void)out_size; (void)d_ws; (void)ws_size;
    const float* vec = (const float*)d_in[0];   // [N,3] float32
    float*       out = (float*)d_out;           // [N,20] float32
    const int n = in_sizes[0] / 3;
    const int blocks = (n + BLOCK - 1) / BLOCK;
    angular_monomials_kernel<<<blocks, BLOCK, 0, stream>>>(vec, out, n);
}